// Probabilistic_SGMlocal_69518340653419
// MI455X (gfx1250) — compile-verified
//
#include <hip/hip_runtime.h>
#include <math.h>

// ---------------------------------------------------------------------------
// Probabilistic SGM local likelihood, fused single-pass per row.
//   out = mean(-0.5*z^2) - LOG_STD - 0.5*LOG2PI,  z = (obs - loc)/STD
// Compute-bound elementwise complex math; db kept in registers; deterministic
// two-kernel reduction (block partials -> 1-wave WMMA-assisted finalize).
// ---------------------------------------------------------------------------

#if defined(__HIP_DEVICE_COMPILE__) && defined(__gfx1250__)
#define CDNA5_PATH 1
#endif

#define S_ROWS 1024
#define F_N    8192
#define TPB    256
#define PER_T  32          // F_N / TPB
#define NWAVES (TPB / 32)

typedef __attribute__((ext_vector_type(2))) float v2f;
typedef __attribute__((ext_vector_type(8))) float v8f;

#ifdef CDNA5_PATH
typedef __attribute__((address_space(1))) int gas_i32;  // global AS
typedef __attribute__((address_space(3))) int las_i32;  // LDS AS
#endif

__global__ __launch_bounds__(TPB)
void sgm_row_kernel(const float* __restrict__ fvec,
                    const float* __restrict__ params,
                    const float* __restrict__ obs,
                    float* __restrict__ partial)
{
    __shared__ float s_obs[F_N];          // 32 KB: obs staged for pass 2
    __shared__ float s_rmin[NWAVES];
    __shared__ float s_rmax[NWAVES];
    __shared__ float s_rsum[NWAVES];

    const int srow = (int)blockIdx.x;
    const int tid  = (int)threadIdx.x;

    // ---- stage obs -> LDS; on CDNA5 use async-to-LDS so the copy overlaps
    //      the pass-1 transcendental work (tracked by ASYNCcnt).
#if defined(CDNA5_PATH) && __has_builtin(__builtin_amdgcn_global_load_async_to_lds_b32)
#define USED_ASYNC_LDS 1
    #pragma unroll
    for (int j = 0; j < PER_T; ++j) {
        const int i = tid + TPB * j;
        __builtin_amdgcn_global_load_async_to_lds_b32(
            (gas_i32*)(obs + i),
            (las_i32*)(&s_obs[i]),
            /*offset=*/0, /*cpol=*/0);
    }
#else
    #pragma unroll
    for (int j = 0; j < PER_T; ++j) {
        const int i = tid + TPB * j;
        s_obs[i] = obs[i];
    }
#endif

    // ---- per-row (wave-uniform) scalars
    const float p0  = params[srow * 4 + 0];
    const float p1  = params[srow * 4 + 1];
    const float p2  = params[srow * 4 + 2];
    const float p3  = params[srow * 4 + 3];
    const float a   = 1000.0f / p0;        // 1/tau_e
    const float b   = 1000.0f / p1;        // 1/tau_i
    const float gb  = p2 * b;              // gii / tau_i
    const float gei = p3;
    const float a2  = a * a, b2 = b * b;
    const float amb = a - b,  ab = a * b;

    // ---- pass 1: db per frequency, kept in registers
    float db_r[PER_T];
    float lmin =  3.402823466e38f;
    float lmax = -3.402823466e38f;
    const float K = 3.0102999566398120f;   // 10 / log2(10)

    #pragma unroll
    for (int j = 0; j < PER_T; ++j) {
        const int   i  = tid + TPB * j;
        const float w  = 6.283185307179586f * fvec[i];
        const float w2 = w * w;
        // Fe = a^2 / (jw + a)^2 = a^2 * ((a2-w2) - 2awi) / (a2+w2)^2
        const float se  = a2 + w2;
        const float si  = b2 + w2;
        const float ke  = a2 / (se * se);
        const float ki  = b2 / (si * si);
        const float Fer = ke * (a2 - w2);
        const float Fei = -2.0f * ke * a * w;
        const float Fir = ki * (b2 - w2);
        const float Fii = -2.0f * ki * b * w;
        // De = jw + a*Fe ; Di = jw + (gii*b)*Fi
        const float Der = a  * Fer,  Dei = w + a  * Fei;
        const float Dir = gb * Fir,  Dii = w + gb * Fii;
        // X = gei * Fe * Fi
        const float Xr = gei * (Fer * Fir - Fei * Fii);
        const float Xi = gei * (Fer * Fii + Fei * Fir);
        // Htotal = (De + Di + (a-b)X) / (De*Di + ab*X^2)
        const float Sr = Der + Dir + amb * Xr;
        const float Si = Dei + Dii + amb * Xi;
        const float Pr = (Der * Dir - Dei * Dii) + ab * (Xr * Xr - Xi * Xi);
        const float Pi = (Der * Dii + Dei * Dir) + ab * (2.0f * Xr * Xi);
        const float ns = Sr * Sr + Si * Si;
        const float nd = Pr * Pr + Pi * Pi;
        const float db = K * (__log2f(ns) - __log2f(nd));  // 20*log10|H|
        db_r[j] = db;
        lmin = fminf(lmin, db);
        lmax = fmaxf(lmax, db);
    }

    // ---- block min/max: wave32 xor-shuffle tree, then 8-slot LDS
    #pragma unroll
    for (int m = 16; m; m >>= 1) {
        lmin = fminf(lmin, __shfl_xor(lmin, m, 32));
        lmax = fmaxf(lmax, __shfl_xor(lmax, m, 32));
    }
    const int wid = tid >> 5;
    if ((tid & 31) == 0) { s_rmin[wid] = lmin; s_rmax[wid] = lmax; }

#if defined(USED_ASYNC_LDS) && __has_builtin(__builtin_amdgcn_s_wait_asynccnt)
    __builtin_amdgcn_s_wait_asynccnt(0);   // obs now resident in LDS
#endif
    __syncthreads();

    float mn = s_rmin[0], mx = s_rmax[0];
    #pragma unroll
    for (int k = 1; k < NWAVES; ++k) {
        mn = fminf(mn, s_rmin[k]);
        mx = fmaxf(mx, s_rmax[k]);
    }

    // ---- pass 2: z^2 accumulation (db in registers, obs in LDS)
    const float inv_rng = 1.0f / (mx - mn);
    const float inv_std = 1.0f / 0.14f;
    float acc = 0.0f;
    #pragma unroll
    for (int j = 0; j < PER_T; ++j) {
        const int   i   = tid + TPB * j;
        const float loc = (db_r[j] - mn) * inv_rng;
        const float z   = (s_obs[i] - loc) * inv_std;
        acc = fmaf(z, z, acc);
    }
    #pragma unroll
    for (int m = 16; m; m >>= 1) acc += __shfl_xor(acc, m, 32);
    if ((tid & 31) == 0) s_rsum[wid] = acc;
    __syncthreads();
    if (tid == 0) {
        float t = 0.0f;
        #pragma unroll
        for (int k = 0; k < NWAVES; ++k) t += s_rsum[k];
        partial[srow] = t;                 // deterministic per-block slot
    }
}

// ---------------------------------------------------------------------------
// Finalize: sum 1024 block partials with one wave. The 32-lane reduction is
// done with v_wmma_f32_16x16x4_f32: A=(v,0) 16x4, B=ones -> D[m][n]=v[m]+v[m+16];
// a lane's 8 D-VGPRs sum to half the total, one xor-16 shuffle completes it.
// ---------------------------------------------------------------------------
__global__ __launch_bounds__(32)
void sgm_final_kernel(const float* __restrict__ partial, float* __restrict__ out)
{
    const int lane = (int)threadIdx.x;
    float s = 0.0f;
    #pragma unroll
    for (int i = 0; i < S_ROWS / 32; ++i) s += partial[lane + 32 * i];

#ifdef CDNA5_PATH
    v2f A; A.x = s;    A.y = 0.0f;   // A[m][0]=v[m], A[m][2]=v[m+16], K=1,3 -> 0
    v2f B; B.x = 1.0f; B.y = 1.0f;   // ones
    v8f C = {};
    // 8 args: (neg_a, A, neg_b, B, c_mod, C, reuse_a, reuse_b)
    C = __builtin_amdgcn_wmma_f32_16x16x4_f32(false, A, false, B,
                                              (short)0, C, false, false);
    float t = ((C[0] + C[1]) + (C[2] + C[3])) + ((C[4] + C[5]) + (C[6] + C[7]));
    t += __shfl_xor(t, 16, 32);      // lanes<16 hold sum(m<8), lanes>=16 sum(m>=8)
#else
    float t = s;
    for (int m = 16; m; m >>= 1) t += __shfl_xor(t, m, 32);
#endif

    if (lane == 0) {
        const float invN = 1.0f / (float)(S_ROWS * F_N);
        // -LOG_STD - 0.5*LOG2PI = -ln(0.14) - 0.5*ln(2*pi)
        const float cst = 1.0471743231681750f;
        out[0] = fmaf(-0.5f * invN, t, cst);
    }
}

extern "C" void kernel_launch(void* const* d_in, const int* in_sizes, int n_in,
                              void* d_out, int out_size, void* d_ws, size_t ws_size,
                              hipStream_t stream)
{
    const float* fvec   = (const float*)d_in[0];   // [F]
    const float* params = (const float*)d_in[1];   // [S,4]
    const float* obs    = (const float*)d_in[2];   // [F]
    float* out     = (float*)d_out;
    float* partial = (float*)d_ws;                 // S_ROWS floats of scratch

    sgm_row_kernel<<<dim3(S_ROWS), dim3(TPB), 0, stream>>>(fvec, params, obs, partial);
    sgm_final_kernel<<<dim3(1), dim3(32), 0, stream>>>(partial, out);

    (void)in_sizes; (void)n_in; (void)out_size; (void)ws_size;
}